// OutHead_72808285602180
// MI455X (gfx1250) — compile-verified
//
#include <hip/hip_runtime.h>
#include <math.h>

// ---------------------------------------------------------------------------
// Equivariant output head, reduced to its live dataflow:
//   LN(x[:, :128]) -> @w1_0/sqrt(128) -> gate -> @w2/sqrt(128)/SCALE
//   -> + shifts[at_no] -> segment_sum over sorted batch_idx
// l=1 / l=2 branches of the reference are dead code (outputs unused).
// GEMM runs on V_WMMA_F32_16X16X4_F32 (exact fp32; kernel is HBM-bound:
// ~67MB live x traffic -> ~3us floor at 23.3 TB/s).
// Round-3 polish: fully unroll the gate/w2 epilogue (kills v_movrels dynamic
// VGPR indexing), explicit unroll hint on the K loop.
// ---------------------------------------------------------------------------

typedef __attribute__((ext_vector_type(2))) float v2f;
typedef __attribute__((ext_vector_type(8))) float v8f;

#define TM 32           // atoms per workgroup (two 16-row WMMA M tiles)
#define XDIM 480        // full feature dim of x
#define L0 128          // live scalar channels
#define APAD 132        // LDS row stride (132 % 64 == 4 -> 16 distinct banks; 16B aligned)

__global__ __launch_bounds__(256) void head_fused_kernel(
    const float* __restrict__ x,        // [N, 480]
    const int*   __restrict__ at_no,    // [N]
    const float* __restrict__ ln_w0,    // [128]
    const float* __restrict__ ln_b0,    // [128]
    const float* __restrict__ w1_0,     // [128, 128] row-major (m, o)
    const float* __restrict__ w2,       // [128, 1]
    const float* __restrict__ shifts,   // [MAX_Z, 1]
    float*       __restrict__ per_atom, // [N] workspace output
    int N)
{
    __shared__ float A[TM][APAD];       // normalized scalar tile
    __shared__ float part[8][TM];       // per-wave partial dots (fixed-order sum)

    const int tid  = threadIdx.x;
    const int wave = tid >> 5;
    const int lane = tid & 31;
    const int l16  = lane & 15;
    const int half = lane >> 4;         // 0: lanes 0-15, 1: lanes 16-31
    const long base = (long)blockIdx.x * TM;

    // ---- stage x[:, :128] for 32 atoms into LDS with 128-bit loads/stores.
    // Row segment is 512B and 16B-aligned (480*4 % 16 == 0).
    for (int i = tid; i < TM * (L0 / 4); i += 256) {
        const int a = i >> 5, c4 = i & 31;           // 32 float4 per row
        const long n = base + a;
        float4 v = make_float4(0.f, 0.f, 0.f, 0.f);
        if (n < N) v = reinterpret_cast<const float4*>(x + n * XDIM)[c4];
        *reinterpret_cast<float4*>(&A[a][c4 * 4]) = v;
    }
    __syncthreads();

    // ---- equivariant LayerNorm on l=0 block: 4 atoms per wave, 4 vals/lane
    #pragma unroll
    for (int s = 0; s < 4; ++s) {
        const int a = wave * 4 + s;
        float v0 = A[a][lane], v1 = A[a][lane + 32];
        float v2 = A[a][lane + 64], v3 = A[a][lane + 96];
        float sum = v0 + v1 + v2 + v3;
        for (int off = 16; off; off >>= 1) sum += __shfl_xor(sum, off, 32);
        const float mean = sum * (1.0f / 128.0f);
        const float c0 = v0 - mean, c1 = v1 - mean;
        const float c2 = v2 - mean, c3 = v3 - mean;
        float sq = c0 * c0 + c1 * c1 + c2 * c2 + c3 * c3;
        for (int off = 16; off; off >>= 1) sq += __shfl_xor(sq, off, 32);
        const float rs = rsqrtf(sq * (1.0f / 128.0f) + 1e-5f);
        A[a][lane]      = c0 * rs * ln_w0[lane]      + ln_b0[lane];
        A[a][lane + 32] = c1 * rs * ln_w0[lane + 32] + ln_b0[lane + 32];
        A[a][lane + 64] = c2 * rs * ln_w0[lane + 64] + ln_b0[lane + 64];
        A[a][lane + 96] = c3 * rs * ln_w0[lane + 96] + ln_b0[lane + 96];
    }
    __syncthreads();

    // ---- GEMM: [32 x 128] @ [128 x 128]; each wave owns a 16-wide N tile
    // and two 16-row M tiles that SHARE each B fragment (1 B load : 2 WMMA).
    // A fragment (16x4 f32): lanes 0-15 -> M=lane, K=k0,k0+1 ; lanes 16-31 -> K=k0+2,k0+3
    // B fragment (4x16 f32): lanes 0-15 -> K=k0 / k0+1 rows ; lanes 16-31 -> K=k0+2 / k0+3
    const int n0 = wave * 16;
    v8f acc0 = {};                       // atoms base+0  .. base+15
    v8f acc1 = {};                       // atoms base+16 .. base+31
    #pragma unroll 8
    for (int kk = 0; kk < 32; ++kk) {
        const int k0 = kk * 4 + half * 2;
        v2f a0, a1, b;
        a0.x = A[l16][k0];       a0.y = A[l16][k0 + 1];
        a1.x = A[16 + l16][k0];  a1.y = A[16 + l16][k0 + 1];
        b.x  = w1_0[(k0)     * 128 + n0 + l16];
        b.y  = w1_0[(k0 + 1) * 128 + n0 + l16];
        acc0 = __builtin_amdgcn_wmma_f32_16x16x4_f32(
            false, a0, false, b, (short)0, acc0, false, false);
        acc1 = __builtin_amdgcn_wmma_f32_16x16x4_f32(
            false, a1, false, b, (short)0, acc1, false, false);
    }

    // ---- fused gate + w2 dot on accumulator fragments (fully unrolled:
    // acc[r] must be static VGPR refs, not m0/v_movrels dynamic indexing).
    // D layout: VGPR r, lanes 0-15 -> (M=r, N=l16); lanes 16-31 -> (M=r+8, N=l16)
    const float inv_sqrt_mul = 0.08838834764831845f;           // 1/sqrt(128)
    const float final_scale  = inv_sqrt_mul / 18.03065905448718f;
    const float w2c = w2[n0 + l16] * final_scale;
    #pragma unroll
    for (int r = 0; r < 8; ++r) {
        float h0 = acc0[r] * inv_sqrt_mul;
        float h1 = acc1[r] * inv_sqrt_mul;
        float g0 = h0 / (1.0f + __expf(-sqrtf(h0 * h0 + 1e-12f)));  // h*sigmoid(|h|)
        float g1 = h1 / (1.0f + __expf(-sqrtf(h1 * h1 + 1e-12f)));
        float val0 = g0 * w2c;
        float val1 = g1 * w2c;
        #pragma unroll
        for (int off = 1; off <= 8; off <<= 1) {                // sum over 16-lane half
            val0 += __shfl_xor(val0, off, 32);
            val1 += __shfl_xor(val1, off, 32);
        }
        if (l16 == 0) {                   // lane0 -> row r, lane16 -> row r+8
            part[wave][r + half * 8]      = val0;
            part[wave][16 + r + half * 8] = val1;
        }
    }
    __syncthreads();

    // ---- deterministic cross-wave combine + per-element shift
    if (tid < TM) {
        const long n = base + tid;
        if (n < N) {
            float s = 0.0f;
            #pragma unroll
            for (int w = 0; w < 8; ++w) s += part[w][tid];
            per_atom[n] = s + shifts[at_no[n]];
        }
    }
}

// ---------------------------------------------------------------------------
// Segment sum over sorted batch_idx: one block per graph, binary-search range,
// fixed-order reduction (deterministic across replays).
// ---------------------------------------------------------------------------
__global__ __launch_bounds__(64) void segsum_kernel(
    const float* __restrict__ per_atom,
    const int*   __restrict__ batch_idx,
    float*       __restrict__ out,
    int N)
{
    const int g = blockIdx.x;
    int lo = 0, hi = N;
    while (lo < hi) { const int m = (lo + hi) >> 1; if (batch_idx[m] <  g) lo = m + 1; else hi = m; }
    const int start = lo;
    hi = N;
    while (lo < hi) { const int m = (lo + hi) >> 1; if (batch_idx[m] <= g) lo = m + 1; else hi = m; }
    const int end = lo;

    float s = 0.0f;
    for (int i = start + threadIdx.x; i < end; i += 64) s += per_atom[i];
    #pragma unroll
    for (int off = 16; off; off >>= 1) s += __shfl_xor(s, off, 32);

    __shared__ float partial[2];
    if ((threadIdx.x & 31) == 0) partial[threadIdx.x >> 5] = s;
    __syncthreads();
    if (threadIdx.x == 0) out[g] = partial[0] + partial[1];
}

extern "C" void kernel_launch(void* const* d_in, const int* in_sizes, int n_in,
                              void* d_out, int out_size, void* d_ws, size_t ws_size,
                              hipStream_t stream) {
    const float* x      = (const float*)d_in[0];
    const int*   at_no  = (const int*)  d_in[1];
    const int*   batch  = (const int*)  d_in[2];
    const float* ln_w0  = (const float*)d_in[3];
    // d_in[4] (ln_w1), d_in[5] (ln_w2): dead paths, unused
    const float* ln_b0  = (const float*)d_in[6];
    const float* w1_0   = (const float*)d_in[7];
    // d_in[8] (w1_1), d_in[9] (w1_2): dead paths, unused
    const float* w2     = (const float*)d_in[10];
    const float* shifts = (const float*)d_in[11];

    const int N = in_sizes[0] / XDIM;
    float* per_atom = (float*)d_ws;     // N floats of scratch

    const int blocks = (N + TM - 1) / TM;
    head_fused_kernel<<<blocks, 256, 0, stream>>>(
        x, at_no, ln_w0, ln_b0, w1_0, w2, shifts, per_atom, N);
    segsum_kernel<<<out_size, 64, 0, stream>>>(per_atom, batch, (float*)d_out, N);
}